// GCN_A_55095840473523
// MI455X (gfx1250) — compile-verified
//
#include <hip/hip_runtime.h>
#include <math.h>

// ---------------- problem constants ----------------
#define BATCH 32
#define NH    778
#define NO    1000
#define HIDD  128
#define NHN   (BATCH*NH)    // 24896
#define NON   (BATCH*NO)    // 32000
#define EH    (BATCH*NH*8)  // 199168
#define EO    (BATCH*NO*8)  // 256000
#define BN_EPS 1e-5f

typedef float v2f __attribute__((ext_vector_type(2)));
typedef float v8f __attribute__((ext_vector_type(8)));

// ---------------- generic fp32 WMMA GEMM ----------------
// C[m,n] (+)= alpha * sum_k A(m,k)*B(k,n) [+ bias[n]] [relu]
// A(m,k) = Ab[m*as_m + k*as_k], B(k,n) = Bb[k*bs_k + n*bs_n], batched via blockIdx.z.
// Fast vectorized staging paths when the tile is fully in-bounds and a stride is 1.
#define TBM 64
#define TBN 32
#define TBK 32
#define APAD 4   // LDS row padding (keeps 16B alignment, kills bank conflicts)

__global__ __launch_bounds__(256)
void wmma_gemm_f32(const float* __restrict__ A, long long as_m, long long as_k, long long as_b,
                   const float* __restrict__ B, long long bs_k, long long bs_n, long long bs_b,
                   float* __restrict__ C, long long cs_b, int ldc,
                   const float* __restrict__ bias,
                   int M, int N, int K, float alpha, int accum, int relu)
{
    __shared__ float As[TBM][TBK + APAD];   // 64*36*4 = 9216 B
    __shared__ float Bs[TBK][TBN + APAD];   // 32*36*4 = 4608 B
    const int b = blockIdx.z;
    const float* Ab = A + (long long)b * as_b;
    const float* Bb = B + (long long)b * bs_b;
    float*       Cb = C + (long long)b * cs_b;

    const int tile_m = blockIdx.y * TBM;
    const int tile_n = blockIdx.x * TBN;
    const int t    = threadIdx.x;
    const int lane = t & 31;
    const int wave = t >> 5;           // 0..7
    const int wm = (wave & 3) * 16;    // wave tile row in block tile
    const int wn = (wave >> 2) * 16;   // wave tile col

    v8f acc = {0.f,0.f,0.f,0.f,0.f,0.f,0.f,0.f};

    const int am    = wm + (lane & 15);
    const int bn    = wn + (lane & 15);
    const int khalf = (lane >> 4) << 1;   // 0 or 2

    const bool mFull = (tile_m + TBM <= M);
    const bool nFull = (tile_n + TBN <= N);

    for (int k0 = 0; k0 < K; k0 += TBK) {
        const bool kFull = (k0 + TBK <= K);

        // ---------- stage A tile (64 x 32) ----------
        if (mFull && kFull && as_k == 1) {
            // row-major A: 512 float4 groups, 2 per thread
            #pragma unroll
            for (int i = 0; i < 2; ++i) {
                int g  = t + i * 256;
                int m  = g >> 3;
                int k4 = (g & 7) << 2;
                float4 v = *(const float4*)(Ab + (long long)(tile_m + m) * as_m + (k0 + k4));
                *(float4*)&As[m][k4] = v;
            }
        } else if (mFull && kFull && as_m == 1) {
            // col-major A (e.g. A_h^T): vectorize along m
            #pragma unroll
            for (int i = 0; i < 2; ++i) {
                int g  = t + i * 256;
                int kk = g & 31;
                int m4 = (g >> 5) << 2;
                float4 v = *(const float4*)(Ab + (tile_m + m4) + (long long)(k0 + kk) * as_k);
                As[m4 + 0][kk] = v.x;
                As[m4 + 1][kk] = v.y;
                As[m4 + 2][kk] = v.z;
                As[m4 + 3][kk] = v.w;
            }
        } else if (mFull && kFull) {
            #pragma unroll
            for (int i = 0; i < 8; ++i) {
                int idx = t + i * 256;
                int m = idx >> 5, kk = idx & 31;
                As[m][kk] = Ab[(long long)(tile_m + m) * as_m + (long long)(k0 + kk) * as_k];
            }
        } else {
            #pragma unroll
            for (int i = 0; i < 8; ++i) {
                int idx = t + i * 256;
                int m = idx >> 5, kk = idx & 31;
                int gm = tile_m + m, gk = k0 + kk;
                As[m][kk] = (gm < M && gk < K)
                          ? Ab[(long long)gm * as_m + (long long)gk * as_k] : 0.f;
            }
        }

        // ---------- stage B tile (32 x 32) ----------
        if (nFull && kFull && bs_n == 1) {
            // 256 float4 groups, 1 per thread
            int kk = t >> 3;
            int n4 = (t & 7) << 2;
            float4 v = *(const float4*)(Bb + (long long)(k0 + kk) * bs_k + (tile_n + n4));
            *(float4*)&Bs[kk][n4] = v;
        } else if (nFull && kFull && bs_k == 1) {
            // B stored n-major (e.g. K_o row-major acting as B^T): vectorize along k
            int n  = t & 31;
            int k4 = (t >> 5) << 2;
            float4 v = *(const float4*)(Bb + (k0 + k4) + (long long)(tile_n + n) * bs_n);
            Bs[k4 + 0][n] = v.x;
            Bs[k4 + 1][n] = v.y;
            Bs[k4 + 2][n] = v.z;
            Bs[k4 + 3][n] = v.w;
        } else if (nFull && kFull) {
            #pragma unroll
            for (int i = 0; i < 4; ++i) {
                int idx = t + i * 256;
                int kk = idx >> 5, n = idx & 31;
                Bs[kk][n] = Bb[(long long)(k0 + kk) * bs_k + (long long)(tile_n + n) * bs_n];
            }
        } else {
            #pragma unroll
            for (int i = 0; i < 4; ++i) {
                int idx = t + i * 256;
                int kk = idx >> 5, n = idx & 31;
                int gk = k0 + kk, gn = tile_n + n;
                Bs[kk][n] = (gk < K && gn < N)
                          ? Bb[(long long)gk * bs_k + (long long)gn * bs_n] : 0.f;
            }
        }
        __syncthreads();

        // ---------- 8 WMMAs over the K tile ----------
        #pragma unroll
        for (int ks = 0; ks < TBK; ks += 4) {
            v2f a = *(const v2f*)&As[am][ks + khalf];
            v2f bb;
            bb.x = Bs[ks + khalf][bn];
            bb.y = Bs[ks + khalf + 1][bn];
            acc = __builtin_amdgcn_wmma_f32_16x16x4_f32(false, a, false, bb,
                                                        (short)0, acc, false, false);
        }
        __syncthreads();
    }

    // C/D layout: VGPR v -> row v (lanes 0-15) / v+8 (lanes 16-31), col = lane&15
    const int row_base = tile_m + wm + ((lane >> 4) << 3);
    const int col      = tile_n + bn;
    if (col < N) {
        const float bv = bias ? bias[col] : 0.f;
        #pragma unroll
        for (int v = 0; v < 8; ++v) {
            int row = row_base + v;
            if (row < M) {
                float r = acc[v] * alpha + bv;
                long long ci = (long long)row * ldc + col;
                if (accum) r += Cb[ci];
                if (relu)  r = fmaxf(r, 0.f);
                Cb[ci] = r;
            }
        }
    }
}

// ---------------- helpers ----------------
__global__ void zero_f32(float* p, long long n) {
    long long i = (long long)blockIdx.x * blockDim.x + threadIdx.x;
    if (i < n) p[i] = 0.f;
}

// scatter for 3-wide features: agg[dst] += x[src]
__global__ void scatter3(const float* __restrict__ x, const int* __restrict__ ei, int E,
                         float* __restrict__ agg) {
    int e = blockIdx.x * blockDim.x + threadIdx.x;
    if (e >= E) return;
    int s = ei[e];
    int d = ei[E + e];
    atomicAdd(&agg[d * 3 + 0], x[s * 3 + 0]);
    atomicAdd(&agg[d * 3 + 1], x[s * 3 + 1]);
    atomicAdd(&agg[d * 3 + 2], x[s * 3 + 2]);
}

// scatter for 128-wide features: one thread handles (edge, 4 channels)
__global__ void scatter128(const float* __restrict__ x, const int* __restrict__ ei, int E,
                           float* __restrict__ agg) {
    long long idx = (long long)blockIdx.x * blockDim.x + threadIdx.x;
    int e  = (int)(idx >> 5);
    int c4 = ((int)idx & 31) << 2;
    if (e >= E) return;
    int s = ei[e];
    int d = ei[E + e];
    const float* xs = x + (long long)s * HIDD + c4;
    float* ad = agg + (long long)d * HIDD + c4;
    atomicAdd(ad + 0, xs[0]);
    atomicAdd(ad + 1, xs[1]);
    atomicAdd(ad + 2, xs[2]);
    atomicAdd(ad + 3, xs[3]);
}

// head GIN MLP: in = x + agg (3 wide) -> relu(lin1 3->256) -> lin2 256->128 -> H
__global__ __launch_bounds__(256)
void head_mlp(const float* __restrict__ x, const float* __restrict__ agg, int n,
              const float* __restrict__ w1, const float* __restrict__ b1,
              const float* __restrict__ w2, const float* __restrict__ b2,
              float* __restrict__ H)
{
    __shared__ float sW1[3 * 256];
    __shared__ float sB1[256];
    __shared__ float sIn[16 * 3];
    __shared__ float sHid[16 * 256];   // 16 KB
    const int t = threadIdx.x;
    for (int i = t; i < 3 * 256; i += 256) sW1[i] = w1[i];
    if (t < 256) sB1[t] = b1[t];
    const int node0 = blockIdx.x * 16;
    if (t < 48) {
        int nd = t / 3, c = t % 3;
        int g = node0 + nd;
        sIn[t] = (g < n) ? (x[g * 3 + c] + agg[g * 3 + c]) : 0.f;
    }
    __syncthreads();
    for (int i = t; i < 16 * 256; i += 256) {
        int nd = i >> 8, j = i & 255;
        float v = sB1[j] + sIn[nd * 3 + 0] * sW1[0 * 256 + j]
                         + sIn[nd * 3 + 1] * sW1[1 * 256 + j]
                         + sIn[nd * 3 + 2] * sW1[2 * 256 + j];
        sHid[i] = fmaxf(v, 0.f);
    }
    __syncthreads();
    for (int i = t; i < 16 * HIDD; i += 256) {
        int nd = i >> 7, c = i & 127;
        int g = node0 + nd;
        if (g < n) {
            float a = b2[c];
            const float* hid = &sHid[nd * 256];
            #pragma unroll 4
            for (int j = 0; j < 256; ++j) a += hid[j] * w2[j * HIDD + c];
            H[(long long)g * HIDD + c] = a;
        }
    }
}

// BN stats: stats[c] += sum, stats[128+c] += sumsq (stats must be pre-zeroed)
__global__ __launch_bounds__(256)
void bn_stats(const float* __restrict__ H, int n, float* __restrict__ stats) {
    __shared__ float ssum[HIDD];
    __shared__ float ssq[HIDD];
    const int t = threadIdx.x;
    if (t < HIDD) { ssum[t] = 0.f; ssq[t] = 0.f; }
    __syncthreads();
    const int r0 = blockIdx.x * 64;
    const int rend = min(r0 + 64, n);
    const int c = t & 127;
    float s = 0.f, q = 0.f;
    for (int r = r0 + (t >> 7); r < rend; r += 2) {
        float v = H[(long long)r * HIDD + c];
        s += v; q += v * v;
    }
    atomicAdd(&ssum[c], s);
    atomicAdd(&ssq[c], q);
    __syncthreads();
    if (t < HIDD) {
        atomicAdd(&stats[t], ssum[t]);
        atomicAdd(&stats[HIDD + t], ssq[t]);
    }
}

// BN apply + relu, in place
__global__ void bn_apply(float* __restrict__ H, int n, const float* __restrict__ stats,
                         const float* __restrict__ g, const float* __restrict__ b) {
    long long i = (long long)blockIdx.x * blockDim.x + threadIdx.x;
    if (i >= (long long)n * HIDD) return;
    int c = (int)(i & 127);
    float inv_n = 1.f / (float)n;
    float mean = stats[c] * inv_n;
    float var  = stats[HIDD + c] * inv_n - mean * mean;
    float v = (H[i] - mean) * rsqrtf(var + BN_EPS) * g[c] + b[c];
    H[i] = fmaxf(v, 0.f);
}

// softmax over contiguous rows (A_o: softmax over axis=2)
__global__ __launch_bounds__(256)
void softmax_rows(float* __restrict__ S, int cols) {
    __shared__ float red[256];
    const long long base = (long long)blockIdx.x * cols;
    const int t = threadIdx.x;
    float m = -1e30f;
    for (int j = t; j < cols; j += 256) m = fmaxf(m, S[base + j]);
    red[t] = m; __syncthreads();
    for (int s = 128; s > 0; s >>= 1) { if (t < s) red[t] = fmaxf(red[t], red[t + s]); __syncthreads(); }
    m = red[0]; __syncthreads();
    float sum = 0.f;
    for (int j = t; j < cols; j += 256) {
        float e = expf(S[base + j] - m);
        S[base + j] = e; sum += e;
    }
    red[t] = sum; __syncthreads();
    for (int s = 128; s > 0; s >>= 1) { if (t < s) red[t] += red[t + s]; __syncthreads(); }
    float inv = 1.f / red[0];
    __syncthreads();
    for (int j = t; j < cols; j += 256) S[base + j] *= inv;
}

// softmax over strided columns (A_h: softmax over hand axis). One wave per (b, o).
__global__ __launch_bounds__(256)
void softmax_cols(float* __restrict__ S, int batches, int Hdim, int Ocols) {
    const int wib  = threadIdx.x >> 5;                  // wave in block (0..7)
    const int lane = threadIdx.x & 31;
    const int gid  = blockIdx.x * 8 + wib;
    const int total = batches * Ocols;
    if (gid >= total) return;
    const int b = gid / Ocols, o = gid - b * Ocols;
    float* base = S + (long long)b * Hdim * Ocols + o;
    float m = -1e30f;
    for (int h = lane; h < Hdim; h += 32) m = fmaxf(m, base[(long long)h * Ocols]);
    for (int off = 16; off > 0; off >>= 1) m = fmaxf(m, __shfl_xor(m, off, 32));
    float sum = 0.f;
    for (int h = lane; h < Hdim; h += 32) {
        float e = expf(base[(long long)h * Ocols] - m);
        base[(long long)h * Ocols] = e; sum += e;
    }
    for (int off = 16; off > 0; off >>= 1) sum += __shfl_xor(sum, off, 32);
    float inv = 1.f / sum;
    for (int h = lane; h < Hdim; h += 32) base[(long long)h * Ocols] *= inv;
}

// tail GIN MLP: in = x + agg (128) -> relu(lin1 128->32) -> lin2 32->dout
__global__ __launch_bounds__(128)
void tail_mlp(const float* __restrict__ xf, const float* __restrict__ agg, int n,
              const float* __restrict__ w1, const float* __restrict__ b1,
              const float* __restrict__ w2, const float* __restrict__ b2,
              float* __restrict__ out, int dout)
{
    __shared__ float sW1[HIDD * 32];   // 16 KB
    __shared__ float sB1[32];
    __shared__ float sW2[32 * 7];
    __shared__ float sB2[7];
    const int t = threadIdx.x;
    for (int i = t; i < HIDD * 32; i += 128) sW1[i] = w1[i];
    if (t < 32) sB1[t] = b1[t];
    for (int i = t; i < 32 * dout; i += 128) sW2[i] = w2[i];
    if (t < dout) sB2[t] = b2[t];
    __syncthreads();
    const int node = blockIdx.x * 128 + t;
    if (node >= n) return;
    float hid[32];
    #pragma unroll
    for (int k = 0; k < 32; ++k) hid[k] = sB1[k];
    const float* xr = xf  + (long long)node * HIDD;
    const float* ar = agg + (long long)node * HIDD;
    for (int j = 0; j < HIDD; ++j) {
        float a = xr[j] + ar[j];
        #pragma unroll
        for (int k = 0; k < 32; ++k) hid[k] += a * sW1[j * 32 + k];
    }
    #pragma unroll
    for (int k = 0; k < 32; ++k) hid[k] = fmaxf(hid[k], 0.f);
    for (int c = 0; c < dout; ++c) {
        float a = sB2[c];
        #pragma unroll
        for (int k = 0; k < 32; ++k) a += hid[k] * sW2[k * dout + c];
        out[(long long)node * dout + c] = a;
    }
}

// ---------------- host side ----------------
static void gemm(hipStream_t s,
                 const float* A, long long as_m, long long as_k, long long as_b,
                 const float* B, long long bs_k, long long bs_n, long long bs_b,
                 float* C, long long cs_b, int ldc, const float* bias,
                 int M, int N, int K, int batches, float alpha, int accum, int relu)
{
    dim3 g((N + TBN - 1) / TBN, (M + TBM - 1) / TBM, batches);
    wmma_gemm_f32<<<g, 256, 0, s>>>(A, as_m, as_k, as_b, B, bs_k, bs_n, bs_b,
                                    C, cs_b, ldc, bias, M, N, K, alpha, accum, relu);
}

static void zero(hipStream_t s, float* p, long long n) {
    zero_f32<<<(unsigned)((n + 255) / 256), 256, 0, s>>>(p, n);
}

extern "C" void kernel_launch(void* const* d_in, const int* in_sizes, int n_in,
                              void* d_out, int out_size, void* d_ws, size_t ws_size,
                              hipStream_t stream)
{
    (void)n_in; (void)out_size; (void)ws_size;
    // Distinguish flattening order: jax-pytree sorted keys vs dict insertion order.
    const bool sorted = (in_sizes[0] == 2 * EH);

    const float *x_h, *x_o;
    const int   *eih, *eio;
    const float *hh_w1, *hh_b1, *hh_w2, *hh_b2, *hh_g, *hh_bb;
    const float *ho_w1, *ho_b1, *ho_w2, *ho_b2, *ho_g, *ho_bb;
    const float *kh_w, *kh_b, *ko_w, *ko_b, *qh_w, *qh_b, *qo_w, *qo_b, *vh_w, *vh_b, *vo_w, *vo_b;
    const float *fh_w, *fh_b, *fo_w, *fo_b;
    const float *th_w1, *th_b1, *th_w2, *th_b2, *to_w1, *to_b1, *to_w2, *to_b2;

    #define FP(i) ((const float*)d_in[(i)])
    #define IP(i) ((const int*)d_in[(i)])
    if (sorted) {
        eih = IP(0); eio = IP(1);
        fh_b = FP(2);  fh_w = FP(3);  fo_b = FP(4);  fo_w = FP(5);
        kh_b = FP(6);  kh_w = FP(7);  ko_b = FP(8);  ko_w = FP(9);
        qh_b = FP(10); qh_w = FP(11); qo_b = FP(12); qo_w = FP(13);
        vh_b = FP(14); vh_w = FP(15); vo_b = FP(16); vo_w = FP(17);
        hh_bb = FP(18); hh_g = FP(19); hh_b1 = FP(20); hh_w1 = FP(21); hh_b2 = FP(22); hh_w2 = FP(23);
        ho_bb = FP(24); ho_g = FP(25); ho_b1 = FP(26); ho_w1 = FP(27); ho_b2 = FP(28); ho_w2 = FP(29);
        // res blocks 30..101 are dead code
        th_b1 = FP(102); th_w1 = FP(103); th_b2 = FP(104); th_w2 = FP(105);
        to_b1 = FP(106); to_w1 = FP(107); to_b2 = FP(108); to_w2 = FP(109);
        x_h = FP(110); x_o = FP(111);
    } else {
        x_h = FP(0); x_o = FP(1); eih = IP(2); eio = IP(3);
        hh_w1 = FP(4);  hh_b1 = FP(5);  hh_w2 = FP(6);  hh_b2 = FP(7);  hh_g = FP(8);  hh_bb = FP(9);
        ho_w1 = FP(10); ho_b1 = FP(11); ho_w2 = FP(12); ho_b2 = FP(13); ho_g = FP(14); ho_bb = FP(15);
        th_w1 = FP(16); th_b1 = FP(17); th_w2 = FP(18); th_b2 = FP(19);
        to_w1 = FP(20); to_b1 = FP(21); to_w2 = FP(22); to_b2 = FP(23);
        // res blocks 24..95 are dead code
        kh_w = FP(96);  kh_b = FP(97);  ko_w = FP(98);  ko_b = FP(99);
        qh_w = FP(100); qh_b = FP(101); qo_w = FP(102); qo_b = FP(103);
        vh_w = FP(104); vh_b = FP(105); vo_w = FP(106); vo_b = FP(107);
        fh_w = FP(108); fh_b = FP(109); fo_w = FP(110); fo_b = FP(111);
    }
    #undef FP
    #undef IP

    // ---- workspace layout (floats) ----
    float* ws = (float*)d_ws;
    long long off = 0;
    float* agg3_h = ws + off; off += (long long)NHN * 3;
    float* agg3_o = ws + off; off += (long long)NON * 3;
    float* stats  = ws + off; off += 512;               // [0..255] hand, [256..511] obj
    float* F_h = ws + off; off += (long long)NHN * HIDD;
    float* F_o = ws + off; off += (long long)NON * HIDD;
    float* Qh  = ws + off; off += (long long)NHN * HIDD;  // later reused as F_ho
    float* Kh  = ws + off; off += (long long)NHN * HIDD;  // later reused as x_h (fused)
    float* Vh  = ws + off; off += (long long)NHN * HIDD;  // later reused as tail agg_h
    float* Qo  = ws + off; off += (long long)NON * HIDD;  // later reused as F_oh
    float* Ko  = ws + off; off += (long long)NON * HIDD;  // later reused as x_o (fused)
    float* Vo  = ws + off; off += (long long)NON * HIDD;  // later reused as tail agg_o

    // ---- output layout ----
    float* out  = (float*)d_out;
    float* outH = out;                                  // 24896*3
    float* outO = out + (long long)NHN * 3;             // 32000*7
    float* AO   = outO + (long long)NON * 7;            // 32*778*1000
    float* AH   = AO + (long long)BATCH * NH * NO;      // 32*778*1000

    const float scale = 1.0f / sqrtf((float)HIDD);

    // 1) GIN aggregation for heads
    zero(stream, agg3_h, (long long)NHN * 3);
    zero(stream, agg3_o, (long long)NON * 3);
    zero(stream, stats, 512);
    scatter3<<<(EH + 255) / 256, 256, 0, stream>>>(x_h, eih, EH, agg3_h);
    scatter3<<<(EO + 255) / 256, 256, 0, stream>>>(x_o, eio, EO, agg3_o);

    // 2) head MLPs -> H in F buffers
    head_mlp<<<(NHN + 15) / 16, 256, 0, stream>>>(x_h, agg3_h, NHN, hh_w1, hh_b1, hh_w2, hh_b2, F_h);
    head_mlp<<<(NON + 15) / 16, 256, 0, stream>>>(x_o, agg3_o, NON, ho_w1, ho_b1, ho_w2, ho_b2, F_o);

    // 3) BatchNorm (training-mode batch stats) + relu
    bn_stats<<<(NHN + 63) / 64, 256, 0, stream>>>(F_h, NHN, stats);
    bn_stats<<<(NON + 63) / 64, 256, 0, stream>>>(F_o, NON, stats + 256);
    bn_apply<<<(unsigned)(((long long)NHN * HIDD + 255) / 256), 256, 0, stream>>>(F_h, NHN, stats, hh_g, hh_bb);
    bn_apply<<<(unsigned)(((long long)NON * HIDD + 255) / 256), 256, 0, stream>>>(F_o, NON, stats + 256, ho_g, ho_bb);

    // 4) K/Q/V projections (WMMA GEMMs, M = total nodes)
    gemm(stream, F_h, HIDD, 1, 0, kh_w, HIDD, 1, 0, Kh, 0, HIDD, kh_b, NHN, HIDD, HIDD, 1, 1.f, 0, 0);
    gemm(stream, F_h, HIDD, 1, 0, qh_w, HIDD, 1, 0, Qh, 0, HIDD, qh_b, NHN, HIDD, HIDD, 1, 1.f, 0, 0);
    gemm(stream, F_h, HIDD, 1, 0, vh_w, HIDD, 1, 0, Vh, 0, HIDD, vh_b, NHN, HIDD, HIDD, 1, 1.f, 0, 0);
    gemm(stream, F_o, HIDD, 1, 0, ko_w, HIDD, 1, 0, Ko, 0, HIDD, ko_b, NON, HIDD, HIDD, 1, 1.f, 0, 0);
    gemm(stream, F_o, HIDD, 1, 0, qo_w, HIDD, 1, 0, Qo, 0, HIDD, qo_b, NON, HIDD, HIDD, 1, 1.f, 0, 0);
    gemm(stream, F_o, HIDD, 1, 0, vo_w, HIDD, 1, 0, Vo, 0, HIDD, vo_b, NON, HIDD, HIDD, 1, 1.f, 0, 0);

    // 5) attention logits (batched): AO[b,h,o] = scale * Qh_b[h] . Ko_b[o]
    gemm(stream, Qh, HIDD, 1, (long long)NH * HIDD,
                 Ko, 1, HIDD, (long long)NO * HIDD,
                 AO, (long long)NH * NO, NO, nullptr, NH, NO, HIDD, BATCH, scale, 0, 0);
    //    AH[b,h,o] = scale * Kh_b[h] . Qo_b[o]
    gemm(stream, Kh, HIDD, 1, (long long)NH * HIDD,
                 Qo, 1, HIDD, (long long)NO * HIDD,
                 AH, (long long)NH * NO, NO, nullptr, NH, NO, HIDD, BATCH, scale, 0, 0);

    // 6) softmax: AO over object axis (rows), AH over hand axis (strided columns)
    softmax_rows<<<BATCH * NH, 256, 0, stream>>>(AO, NO);
    softmax_cols<<<(BATCH * NO + 7) / 8, 256, 0, stream>>>(AH, BATCH, NH, NO);

    // 7) attention apply: F_ho = AO @ Vo (into Qh), F_oh = AH^T @ Vh (into Qo)
    gemm(stream, AO, NO, 1, (long long)NH * NO,
                 Vo, HIDD, 1, (long long)NO * HIDD,
                 Qh, (long long)NH * HIDD, HIDD, nullptr, NH, HIDD, NO, BATCH, 1.f, 0, 0);
    gemm(stream, AH, 1, NO, (long long)NH * NO,
                 Vh, HIDD, 1, (long long)NH * HIDD,
                 Qo, (long long)NO * HIDD, HIDD, nullptr, NO, HIDD, NH, BATCH, 1.f, 0, 0);

    // 8) fuse linears: x = [F, F_attn] @ W(256x128) + b   (two passes, second accumulates)
    gemm(stream, F_h, HIDD, 1, 0, fh_w, HIDD, 1, 0, Kh, 0, HIDD, fh_b, NHN, HIDD, HIDD, 1, 1.f, 0, 0);
    gemm(stream, Qh,  HIDD, 1, 0, fh_w + HIDD * HIDD, HIDD, 1, 0, Kh, 0, HIDD, nullptr, NHN, HIDD, HIDD, 1, 1.f, 1, 0);
    gemm(stream, F_o, HIDD, 1, 0, fo_w, HIDD, 1, 0, Ko, 0, HIDD, fo_b, NON, HIDD, HIDD, 1, 1.f, 0, 0);
    gemm(stream, Qo,  HIDD, 1, 0, fo_w + HIDD * HIDD, HIDD, 1, 0, Ko, 0, HIDD, nullptr, NON, HIDD, HIDD, 1, 1.f, 1, 0);

    // 9) tail GIN: aggregate fused features, then 128->32->dout MLP
    zero(stream, Vh, (long long)NHN * HIDD);
    zero(stream, Vo, (long long)NON * HIDD);
    scatter128<<<(unsigned)(((long long)EH * 32 + 255) / 256), 256, 0, stream>>>(Kh, eih, EH, Vh);
    scatter128<<<(unsigned)(((long long)EO * 32 + 255) / 256), 256, 0, stream>>>(Ko, eio, EO, Vo);
    tail_mlp<<<(NHN + 127) / 128, 128, 0, stream>>>(Kh, Vh, NHN, th_w1, th_b1, th_w2, th_b2, outH, 3);
    tail_mlp<<<(NON + 127) / 128, 128, 0, stream>>>(Ko, Vo, NON, to_w1, to_b1, to_w2, to_b2, outO, 7);
}